// Sort_59949153517723
// MI455X (gfx1250) — compile-verified
//
#include <hip/hip_runtime.h>
#include <cstdint>
#include <cstddef>

// Problem shape (fixed by the reference): x is (32, 4096, 256) f32.
// out[b, p, :] = x[b, idx[b][p], :] where idx = stable argsort of -x[:, :, 5].
constexpr int kB = 32;
constexpr int kN = 4096;
constexpr int kC = 256;
constexpr int kColIdx = 5;

constexpr int kSortThreads = 1024;           // 32 wave32 waves, max workgroup
constexpr int kEPT = kN / kSortThreads;      // elements per thread = 4
static_assert(kN % kSortThreads == 0, "");

// gfx1250 async global->LDS path (compile-probed via __has_builtin; plain-load
// fallback otherwise so this file always compiles). Probe round 1 confirmed
// the builtin exists here and takes (as1 int*, as3 int*, imm, imm).
#if defined(__HIP_DEVICE_COMPILE__) &&                                        \
    __has_builtin(__builtin_amdgcn_global_load_async_to_lds_b32) &&           \
    __has_builtin(__builtin_amdgcn_s_wait_asynccnt)
#define USE_ASYNC_LDS 1
#else
#define USE_ASYNC_LDS 0
#endif

#if USE_ASYNC_LDS
typedef __attribute__((address_space(1))) int g_int;
typedef __attribute__((address_space(3))) int l_int;
#endif

// ---------------------------------------------------------------------------
// Kernel 1: per-batch stable argsort of keys = x[b, :, 5], descending.
// One workgroup per batch; bitonic sort of (key, idx) pairs in LDS.
// Comparator: a before b  <=>  key_a > key_b, ties broken by idx_a < idx_b.
// This total order makes bitonic sort reproduce jnp.argsort(-keys) (stable).
// ---------------------------------------------------------------------------
__global__ __launch_bounds__(kSortThreads) void argsort_keys_kernel(
    const float* __restrict__ x, int* __restrict__ perm)
{
  __shared__ float sk[kN];   // 16 KB
  __shared__ int   si[kN];   // 16 KB
  const int b   = blockIdx.x;
  const int tid = threadIdx.x;
  const float* __restrict__ xb = x + (size_t)b * kN * kC;

#if USE_ASYNC_LDS
  // Stage strided keys straight into LDS with the CDNA5 async data mover path;
  // no VGPR round-trip, tracked on ASYNCcnt.
  for (int m = 0; m < kEPT; ++m) {
    const int i = tid + m * kSortThreads;
    __builtin_amdgcn_global_load_async_to_lds_b32(
        (g_int*)(uintptr_t)(xb + (size_t)i * kC + kColIdx),
        (l_int*)(uint32_t)(uintptr_t)(&sk[i]),
        /*imm offset*/ 0, /*cpol*/ 0);
    si[i] = i;
  }
  __builtin_amdgcn_s_wait_asynccnt(0);
#else
  for (int m = 0; m < kEPT; ++m) {
    const int i = tid + m * kSortThreads;
    sk[i] = xb[(size_t)i * kC + kColIdx];
    si[i] = i;
  }
#endif
  __syncthreads();

  for (int k = 2; k <= kN; k <<= 1) {
    for (int j = k >> 1; j > 0; j >>= 1) {
#pragma unroll
      for (int m = 0; m < kEPT; ++m) {
        const int i = tid + m * kSortThreads;
        const int l = i ^ j;
        if (l > i) {  // each disjoint pair handled exactly once per step
          const float ka = sk[i], kb = sk[l];
          const int   ia = si[i], ib = si[l];
          const bool a_before_b = (ka > kb) || ((ka == kb) && (ia < ib));
          const bool up = ((i & k) == 0);
          if (up ? !a_before_b : a_before_b) {
            sk[i] = kb; sk[l] = ka;
            si[i] = ib; si[l] = ia;
          }
        }
      }
      __syncthreads();
    }
  }

  for (int m = 0; m < kEPT; ++m) {
    const int i = tid + m * kSortThreads;
    perm[b * kN + i] = si[i];
  }
}

// ---------------------------------------------------------------------------
// Kernel 2: bandwidth-bound row gather. 4 rows per 256-thread block; each lane
// moves one float4 (global_load_b128 / global_store_b128). Rows are 1 KB
// contiguous, so loads and stores are fully coalesced for any permutation.
// ---------------------------------------------------------------------------
__global__ __launch_bounds__(256) void gather_rows_kernel(
    const float* __restrict__ x, const int* __restrict__ perm,
    float* __restrict__ out)
{
  const int b    = blockIdx.y;
  const int row  = blockIdx.x * 4 + (threadIdx.x >> 6);   // dest row
  const int col  = (threadIdx.x & 63) << 2;               // 64 lanes * float4
  const int src  = perm[b * kN + row];

  const float4* __restrict__ in =
      (const float4*)(x + ((size_t)b * kN + src) * kC + col);
  float4* __restrict__ o =
      (float4*)(out + ((size_t)b * kN + row) * kC + col);
  *o = *in;
}

// ---------------------------------------------------------------------------
extern "C" void kernel_launch(void* const* d_in, const int* in_sizes, int n_in,
                              void* d_out, int out_size, void* d_ws,
                              size_t ws_size, hipStream_t stream)
{
  (void)in_sizes; (void)n_in; (void)out_size; (void)ws_size;
  const float* x   = (const float*)d_in[0];
  float*       out = (float*)d_out;
  int*         perm = (int*)d_ws;   // 32 * 4096 * 4 B = 512 KB scratch

  argsort_keys_kernel<<<dim3(kB), dim3(kSortThreads), 0, stream>>>(x, perm);
  gather_rows_kernel<<<dim3(kN / 4, kB), dim3(256), 0, stream>>>(x, perm, out);
}